// DGCNN_83030307766890
// MI455X (gfx1250) — compile-verified
//
#include <hip/hip_runtime.h>
#include <math.h>

// DGCNN forward for MI455X (gfx1250): one workgroup per graph, whole graph
// resident in LDS (320KB/WGP), dense X@W tiles via V_WMMA_F32_16X16X4_F32,
// edge aggregation via ds_add_f32 LDS atomics.

typedef float v2f __attribute__((ext_vector_type(2)));
typedef float v8f __attribute__((ext_vector_type(8)));

#define NGRAPH 512
#define MNODES 200
#define EPG    6400            // edges per graph (E/B)
#define ETOT   (NGRAPH * EPG)  // 3,276,800
#define ROWS   208             // 200 padded to 13*16 for WMMA M-tiles
#define HSTR   36              // LDS row stride (floats): 144B, 16B aligned, conflict-free
#define NTILES 26              // 13 M-tiles * 2 N-tiles (32 output features)

// ---- shared memory layout (floats) ----
#define OFF_XW    0
#define OFF_AGG   (OFF_XW   + ROWS*HSTR)
#define OFF_H1    (OFF_AGG  + ROWS*HSTR)
#define OFF_H2    (OFF_H1   + ROWS*HSTR)
#define OFF_H3    (OFF_H2   + ROWS*HSTR)
#define OFF_DINV  (OFF_H3   + ROWS*HSTR)
#define OFF_H4    (OFF_DINV + ROWS)
#define OFF_XW4   (OFF_H4   + ROWS)
#define OFF_AGG4  (OFF_XW4  + ROWS)
#define OFF_DEGI  (OFF_AGG4 + ROWS)   // int[ROWS]
#define OFF_TOP   (OFF_DEGI + ROWS)   // int[32]
#define OFF_C1    (OFF_TOP  + 32)     // [30][16]
#define OFF_P     (OFF_C1   + 480)    // [16][15]
#define OFF_C2    (OFF_P    + 240)    // [352]
#define OFF_HLIN  (OFF_C2   + 352)    // [128]
#define SMEM_FLOATS (OFF_HLIN + 128)  // 39712 floats = 158,848 bytes

// Dense GEMM: out[ROWS][32] = A[200..pad][K] @ W[K][32] with f32 WMMA 16x16x4.
// FROM_LDS=false: A rows come from global (row-major, K cols), rows>=200 masked 0.
// FROM_LDS=true : A comes from LDS with stride HSTR (padded rows already zero).
template <bool FROM_LDS>
__device__ __forceinline__ void gemm_xw(const float* __restrict__ Ag,
                                        const float* __restrict__ Alds,
                                        const float* __restrict__ W, int K,
                                        float* __restrict__ out,
                                        int lane, int wave) {
  const int hi  = lane >> 4;      // 0: lanes 0-15, 1: lanes 16-31
  const int l15 = lane & 15;
  for (int t = wave; t < NTILES; t += 8) {
    const int tm = t % 13, tn = t / 13;
    const int rowA = tm * 16 + l15;          // M index this lane supplies
    const int n    = (tn << 4) + l15;        // N index this lane holds in B/C/D
    v8f c = {};
    for (int kb = 0; kb < K; kb += 4) {
      const int ka = kb + (hi << 1);         // lanes 16-31 hold K+2, K+3
      v2f a, b;
      if constexpr (FROM_LDS) {
        a.x = Alds[rowA * HSTR + ka];
        a.y = Alds[rowA * HSTR + ka + 1];
      } else {
        const float msk = (rowA < MNODES) ? 1.0f : 0.0f;
        const int   rA  = (rowA < MNODES) ? rowA : 0;
        const float* ap = Ag + (size_t)rA * K + ka;
        a.x = ap[0] * msk;
        a.y = ap[1] * msk;
      }
      b.x = W[(ka + 0) * 32 + n];
      b.y = W[(ka + 1) * 32 + n];
      // D = A*B + C, full f32 (16x16x4)
      c = __builtin_amdgcn_wmma_f32_16x16x4_f32(false, a, false, b,
                                                (short)0, c, false, false);
    }
    // C/D layout: VGPR r -> M = tm*16 + 8*hi + r, N = n
#pragma unroll
    for (int r = 0; r < 8; ++r)
      out[(tm * 16 + (hi << 3) + r) * HSTR + n] = c[r];
  }
}

// scatter agg[dst] += xw[src]*norm over this graph's edges, then
// h = tanh(agg + xw*dinv^2 + bias) into hout (padded rows zeroed).
__device__ __forceinline__ void aggregate_combine(
    const int* __restrict__ srcA, const int* __restrict__ dstA,
    int eBase, int nodeBase, const float* __restrict__ dinv,
    const float* __restrict__ xw, float* __restrict__ agg,
    const float* __restrict__ bias, float* __restrict__ hout, int tid) {
  for (int idx = tid; idx < ROWS * 32; idx += 256)
    agg[(idx >> 5) * HSTR + (idx & 31)] = 0.0f;
  __syncthreads();   // also orders prior WMMA writes to xw before reads below
  for (int e0 = 0; e0 < EPG; e0 += 32) {
    const int e = eBase + e0 + (tid >> 3);
    const int f = (tid & 7) << 2;
    const int s = srcA[e] - nodeBase;
    const int d = dstA[e] - nodeBase;
    const float nm = dinv[s] * dinv[d];
    const float4 v = *(const float4*)(xw + s * HSTR + f);
    float* ap = agg + d * HSTR + f;
    atomicAdd(ap + 0, v.x * nm);
    atomicAdd(ap + 1, v.y * nm);
    atomicAdd(ap + 2, v.z * nm);
    atomicAdd(ap + 3, v.w * nm);
  }
  __syncthreads();
  for (int idx = tid; idx < ROWS * 32; idx += 256) {
    const int i = idx >> 5, f = idx & 31;
    float val = 0.0f;
    if (i < MNODES) {
      const float di = dinv[i];
      val = tanhf(agg[i * HSTR + f] + xw[i * HSTR + f] * di * di + bias[f]);
    }
    hout[i * HSTR + f] = val;
  }
  __syncthreads();
}

__global__ __launch_bounds__(256) void dgcnn_fused_kernel(
    const float* __restrict__ x, const int* __restrict__ eidx,
    const float* __restrict__ W1, const float* __restrict__ b1,
    const float* __restrict__ W2, const float* __restrict__ b2,
    const float* __restrict__ W3, const float* __restrict__ b3,
    const float* __restrict__ W4, const float* __restrict__ b4,
    const float* __restrict__ cW1, const float* __restrict__ cb1,
    const float* __restrict__ cW2, const float* __restrict__ cb2,
    const float* __restrict__ lW1, const float* __restrict__ lb1,
    const float* __restrict__ lW2, const float* __restrict__ lb2,
    float* __restrict__ out) {
  extern __shared__ __align__(16) float smem[];
  const int g    = blockIdx.x;
  const int tid  = threadIdx.x;
  const int lane = tid & 31;
  const int wave = tid >> 5;

  const int nodeBase = g * MNODES;
  const size_t xBase = (size_t)nodeBase * 128;
  const int* srcA = eidx;
  const int* dstA = eidx + ETOT;
  const int eBase = g * EPG;

  float* xw   = smem + OFF_XW;
  float* agg  = smem + OFF_AGG;
  float* h1   = smem + OFF_H1;
  float* h2   = smem + OFF_H2;
  float* h3   = smem + OFF_H3;
  float* dinv = smem + OFF_DINV;
  float* h4v  = smem + OFF_H4;
  float* xw4  = smem + OFF_XW4;
  float* agg4 = smem + OFF_AGG4;
  int*   degi = (int*)(smem + OFF_DEGI);
  int*   topIdx = (int*)(smem + OFF_TOP);
  float* c1buf  = smem + OFF_C1;
  float* pbuf   = smem + OFF_P;
  float* c2buf  = smem + OFF_C2;
  float* hlin   = smem + OFF_HLIN;

  // warm L2/WGP$ with this graph's x block (emits global_prefetch_b8)
  if (tid < MNODES) __builtin_prefetch(x + xBase + (size_t)tid * 128, 0, 1);

  // ---- degrees + dinv ----
  for (int i = tid; i < ROWS; i += 256) degi[i] = 0;
  __syncthreads();
  for (int e = tid; e < EPG; e += 256)
    atomicAdd(&degi[dstA[eBase + e] - nodeBase], 1);
  __syncthreads();
  for (int i = tid; i < ROWS; i += 256)
    dinv[i] = (i < MNODES) ? rsqrtf((float)degi[i] + 1.0f) : 0.0f;
  __syncthreads();

  // ---- GCN layers 1..3 (WMMA GEMM + LDS-atomic scatter) ----
  gemm_xw<false>(x + xBase, nullptr, W1, 128, xw, lane, wave);
  aggregate_combine(srcA, dstA, eBase, nodeBase, dinv, xw, agg, b1, h1, tid);

  gemm_xw<true>(nullptr, h1, W2, 32, xw, lane, wave);
  aggregate_combine(srcA, dstA, eBase, nodeBase, dinv, xw, agg, b2, h2, tid);

  gemm_xw<true>(nullptr, h2, W3, 32, xw, lane, wave);
  aggregate_combine(srcA, dstA, eBase, nodeBase, dinv, xw, agg, b3, h3, tid);

  // ---- layer 4: 32 -> 1 ----
  for (int i = tid; i < ROWS; i += 256) {
    float s = 0.0f;
    if (i < MNODES) {
#pragma unroll
      for (int k = 0; k < 32; ++k) s += h3[i * HSTR + k] * W4[k];
    }
    xw4[i] = s;
    agg4[i] = 0.0f;
  }
  __syncthreads();
  for (int e = tid; e < EPG; e += 256) {
    const int s = srcA[eBase + e] - nodeBase;
    const int d = dstA[eBase + e] - nodeBase;
    atomicAdd(&agg4[d], xw4[s] * dinv[s] * dinv[d]);
  }
  __syncthreads();
  for (int i = tid; i < ROWS; i += 256) {
    const float di = dinv[i];
    h4v[i] = (i < MNODES) ? tanhf(agg4[i] + xw4[i] * di * di + b4[0]) : 0.0f;
  }
  __syncthreads();

  // ---- SortPool: rank by h4 desc, stable (index asc) -> top 30 ----
  if (tid < MNODES) {
    const float vi = h4v[tid];
    int cnt = 0;
    for (int j = 0; j < MNODES; ++j) {
      const float vj = h4v[j];
      cnt += (vj > vi) || (vj == vi && j < tid);
    }
    if (cnt < 30) topIdx[cnt] = tid;
  }
  __syncthreads();

  // ---- conv1: [30,97] @ [97,16]^T, relu ----
  for (int idx = tid; idx < 480; idx += 256) {
    const int k = idx >> 4, o = idx & 15;
    const int nd = topIdx[k];
    const float* wrow = cW1 + o * 97;
    float s = cb1[o];
#pragma unroll 8
    for (int f = 0; f < 32; ++f) s += h1[nd * HSTR + f] * wrow[f];
#pragma unroll 8
    for (int f = 0; f < 32; ++f) s += h2[nd * HSTR + f] * wrow[32 + f];
#pragma unroll 8
    for (int f = 0; f < 32; ++f) s += h3[nd * HSTR + f] * wrow[64 + f];
    s += h4v[nd] * wrow[96];
    c1buf[idx] = fmaxf(s, 0.0f);   // layout [k][o]
  }
  __syncthreads();

  // ---- maxpool(2,2) -> p[16][15] ----
  for (int idx = tid; idx < 240; idx += 256) {
    const int o = idx / 15, k2 = idx % 15;
    pbuf[o * 15 + k2] =
        fmaxf(c1buf[(2 * k2) * 16 + o], c1buf[(2 * k2 + 1) * 16 + o]);
  }
  __syncthreads();

  // ---- conv2: [32,16,5] over p -> c2[32][11], relu ----
  for (int idx = tid; idx < 352; idx += 256) {
    const int o2 = idx / 11, t = idx % 11;
    const float* w = cW2 + o2 * 80;  // [16][5]
    float s = cb2[o2];
#pragma unroll
    for (int i = 0; i < 16; ++i)
#pragma unroll
      for (int r = 0; r < 5; ++r) s += pbuf[i * 15 + t + r] * w[i * 5 + r];
    c2buf[idx] = fmaxf(s, 0.0f);     // flat = [o2*11 + t]
  }
  __syncthreads();

  // ---- lin1: 352 -> 128, relu ----
  for (int j = tid; j < 128; j += 256) {
    float s = lb1[j];
    for (int i = 0; i < 352; ++i) s += c2buf[i] * lW1[i * 128 + j];
    hlin[j] = fmaxf(s, 0.0f);
  }
  __syncthreads();

  // ---- lin2 + sigmoid ----
  if (tid == 0) {
    float s = lb2[0];
#pragma unroll 16
    for (int j = 0; j < 128; ++j) s += hlin[j] * lW2[j];
    out[g] = 1.0f / (1.0f + expf(-s));
  }
}

extern "C" void kernel_launch(void* const* d_in, const int* in_sizes, int n_in,
                              void* d_out, int out_size, void* d_ws,
                              size_t ws_size, hipStream_t stream) {
  (void)in_sizes; (void)n_in; (void)d_ws; (void)ws_size; (void)out_size;
  const float* x    = (const float*)d_in[0];
  const int*   eidx = (const int*)d_in[1];
  // d_in[2] = batch (unused: block-contiguous by construction)
  const float* W1 = (const float*)d_in[3];
  const float* b1 = (const float*)d_in[4];
  const float* W2 = (const float*)d_in[5];
  const float* b2 = (const float*)d_in[6];
  const float* W3 = (const float*)d_in[7];
  const float* b3 = (const float*)d_in[8];
  const float* W4 = (const float*)d_in[9];
  const float* b4 = (const float*)d_in[10];
  const float* cW1 = (const float*)d_in[11];
  const float* cb1 = (const float*)d_in[12];
  const float* cW2 = (const float*)d_in[13];
  const float* cb2 = (const float*)d_in[14];
  const float* lW1 = (const float*)d_in[15];
  const float* lb1 = (const float*)d_in[16];
  const float* lW2 = (const float*)d_in[17];
  const float* lb2 = (const float*)d_in[18];
  float* out = (float*)d_out;

  const size_t smemBytes = (size_t)SMEM_FLOATS * sizeof(float);
  dgcnn_fused_kernel<<<dim3(NGRAPH), dim3(256), smemBytes, stream>>>(
      x, eidx, W1, b1, W2, b2, W3, b3, W4, b4, cW1, cb1, cW2, cb2, lW1, lb1,
      lW2, lb2, out);
}